// BSplineBasis_11836929868211
// MI455X (gfx1250) — compile-verified
//
#include <hip/hip_runtime.h>

// B-spline basis (Cox-de Boor, uniform knots, cubic) on MI455X / gfx1250.
//
// Roofline: out = 8.39M x 8 fp32 = 256MiB, in = 32MiB  ->  ~288MiB of HBM
// traffic -> ~12.4us floor at 23.3 TB/s. Pure bandwidth problem; the job of
// the kernel is to keep VALU cost/element tiny and stores wide.
//
// Formulation: uniform knots => per element only 4 of the 8 outputs are
// nonzero: out[i-3+q] = P_q(u), the standard uniform cubic B-spline pieces.
// Written as out(8) = feat(20) x M(20x8) with feat = onehot(interval,5) (x)
// [1,u,u^2,u^3], this is a per-16-element GEMM on V_WMMA_F32_16X16X4_F32,
// accumulated over 5 K-blocks:
//   A (16x4)  = constant coefficient fragment (M = output col, K = power)
//   B (4x16)  = per-element features          (N = element)
//   D (16x16) = rows 0..7 = the 8 outputs, column n = element n
// D layout puts all 8 outputs of element n in VGPRs 0..7 of lane n, so each
// of lanes 0..15 does two global_store_b128 -> wide contiguous stores.

typedef __attribute__((ext_vector_type(2))) float v2f;
typedef __attribute__((ext_vector_type(4))) float v4f;
typedef __attribute__((ext_vector_type(8))) float v8f;

#define ORDER   3
#define NBLK    5   // grid_size (number of interior intervals / K-blocks)
#define NOUT    8   // grid_size + order

__global__ __launch_bounds__(256)
void bspline_wmma_kernel(const float* __restrict__ X,
                         const float* __restrict__ G,
                         float* __restrict__ Out,
                         int nTiles, int rem)
{
    const int lane = threadIdx.x & 31;
    const int el   = lane & 15;        // element slot within the 16-wide tile
    const int half = (lane >> 4) & 1;  // 0 -> K{0,1}, 1 -> K{2,3} (f32 WMMA layout)

    // Uniform grid parameters (grid is linspace by construction).
    const float g0    = G[0];
    const float inv_h = 1.0f / (G[1] - G[0]);

    // Uniform cubic B-spline piece coefficients over powers [1,u,u^2,u^3], /6.
    const float S = 1.0f / 6.0f;
    const float C0[4] = { 1.f*S, -3.f*S,  3.f*S, -1.f*S };  // (1-u)^3/6
    const float C1[4] = { 4.f*S,  0.f  , -6.f*S,  3.f*S };  // (3u^3-6u^2+4)/6
    const float C2[4] = { 1.f*S,  3.f*S,  3.f*S, -3.f*S };  // (-3u^3+3u^2+3u+1)/6
    const float C3[4] = { 0.f  ,  0.f  ,  0.f  ,  1.f*S };  // u^3/6

    // Build per-lane A fragments for the 5 interval blocks, once.
    // A is 16x4: row M = lane&15 (output column), K pair selected by wave half.
    // Rows >= NOUT are zero so D rows 8..15 (lanes 16..31) stay zero.
    v2f A[NBLK];
    {
        const int n = el;
        #pragma unroll
        for (int m = 0; m < NBLK; ++m) {
            float a0 = 0.f, a1 = 0.f;
            #pragma unroll
            for (int q = 0; q < 4; ++q) {
                const float* Cq = (q == 0) ? C0 : (q == 1) ? C1 : (q == 2) ? C2 : C3;
                bool pick = ((n - m) == q) && (n < NOUT);
                float c0 = half ? Cq[2] : Cq[0];
                float c1 = half ? Cq[3] : Cq[1];
                a0 = pick ? c0 : a0;
                a1 = pick ? c1 : a1;
            }
            A[m].x = a0;
            A[m].y = a1;
        }
    }

    const int wavesTotal = (int)((gridDim.x * blockDim.x) >> 5);
    const int wave       = (int)((blockIdx.x * blockDim.x + threadIdx.x) >> 5);

    // Wave-uniform tile loop: EXEC stays all-ones across the WMMA ops.
    for (int tile = wave; tile < nTiles; tile += wavesTotal) {
        const long long e = (long long)tile * 16 + el;
        float x = X[e];
        // Prefetch next tile's x (gfx1250 global_prefetch_b8).
        __builtin_prefetch(&X[e + (long long)wavesTotal * 16], 0, 1);

        // Interval + local coordinate.
        float t = (x - g0) * inv_h;
        int   i = (int)floorf(t);
        i = min(max(i, ORDER), ORDER + NBLK - 1);   // valid support: [3, 7]
        float u  = t - (float)i;
        float u2 = u * u;
        float u3 = u2 * u;
        // This lane's K-pair of the feature vector [1, u, u^2, u^3].
        float f0 = half ? u2 : 1.0f;
        float f1 = half ? u3 : u;

        v8f acc = {};
        #pragma unroll
        for (int m = 0; m < NBLK; ++m) {
            bool s = (i == m + ORDER);        // onehot over interval blocks
            v2f b;
            b.x = s ? f0 : 0.0f;
            b.y = s ? f1 : 0.0f;
            // acc(16x16) += A_m(16x4) x B_m(4x16)
            acc = __builtin_amdgcn_wmma_f32_16x16x4_f32(
                /*neg_a=*/false, A[m], /*neg_b=*/false, b,
                /*c_mod=*/(short)0, acc, /*reuse_a=*/false, /*reuse_b=*/false);
        }

        // Lane n (< 16) holds outputs 0..7 of element tile*16+n in acc[0..7].
        if (lane < 16) {
            float* o = Out + e * NOUT;
            v4f lo = { acc[0], acc[1], acc[2], acc[3] };
            v4f hi = { acc[4], acc[5], acc[6], acc[7] };
            *(v4f*)(o)     = lo;
            *(v4f*)(o + 4) = hi;
        }
    }

    // Scalar tail for N % 16 elements (closed form, same math).
    if (rem > 0) {
        int gid = (int)(blockIdx.x * blockDim.x + threadIdx.x);
        if (gid < rem) {
            long long e = (long long)nTiles * 16 + gid;
            float x = X[e];
            float t = (x - g0) * inv_h;
            int   i = min(max((int)floorf(t), ORDER), ORDER + NBLK - 1);
            float u  = t - (float)i;
            float om = 1.0f - u;
            float u2 = u * u, u3 = u2 * u;
            float w0 = om * om * om * S;
            float w1 = (3.f * u3 - 6.f * u2 + 4.f) * S;
            float w2 = (-3.f * u3 + 3.f * u2 + 3.f * u + 1.f) * S;
            float w3 = u3 * S;
            float* o = Out + e * NOUT;
            #pragma unroll
            for (int j = 0; j < NOUT; ++j) {
                int q = j - (i - ORDER);
                float v = 0.f;
                v = (q == 0) ? w0 : v;
                v = (q == 1) ? w1 : v;
                v = (q == 2) ? w2 : v;
                v = (q == 3) ? w3 : v;
                o[j] = v;
            }
        }
    }
}

extern "C" void kernel_launch(void* const* d_in, const int* in_sizes, int n_in,
                              void* d_out, int out_size, void* d_ws, size_t ws_size,
                              hipStream_t stream) {
    const float* X = (const float*)d_in[0];   // (2048, 4096) fp32
    const float* G = (const float*)d_in[1];   // (12,) fp32
    float* Out = (float*)d_out;               // (2048, 4096, 8) fp32

    long long N = (long long)in_sizes[0];
    int nTiles = (int)(N / 16);
    int rem    = (int)(N - (long long)nTiles * 16);

    // 2048 blocks x 256 threads = 16384 waves; grid-stride over 524288 tiles.
    dim3 grid(2048), block(256);
    hipLaunchKernelGGL(bspline_wmma_kernel, grid, block, 0, stream,
                       X, G, Out, nTiles, rem);
}